// EdgeConvModel_11407433138819
// MI455X (gfx1250) — compile-verified
//
#include <hip/hip_runtime.h>
#include <hip/hip_bf16.h>
#include <math.h>

typedef __attribute__((ext_vector_type(2))) float v2f;
typedef __attribute__((ext_vector_type(8))) float v8f;

#define BN_EPS 1e-3f

// Fragment-major packing for V_WMMA_F32_16X16X4_F32 operands.
// Lane l = half*16 + m consumes elements (k = 4t + 2*half + j, m), j=0,1.
// Stored at dword  t*64 + half*32 + m*2 + j  == t*64 + lane*2 + j,
// so each operand is one aligned ds_load_b64, banks lane*2..lane*2+1
// (all 64 banks, conflict-free).
__device__ __forceinline__ int fragIdx(int k, int m) {
    return ((k >> 2) << 6) + (((k >> 1) & 1) << 5) + (m << 1) + (k & 1);
}

// ---------------------------------------------------------------------------
// Phase 0: zero workspace (agg[N*64] + deg[N])
// ---------------------------------------------------------------------------
__global__ void zero_ws_kernel(float* __restrict__ p, long long n) {
    long long i = (long long)blockIdx.x * blockDim.x + threadIdx.x;
    if (i < n) p[i] = 0.0f;
}

// ---------------------------------------------------------------------------
// Phase 1: scatter. 16 threads per edge; each does a float4 read of the
// sender row (L2-resident, coalesced) and 4 global_atomic_add_f32 into
// agg[receiver]. Thread q==0 also bumps deg[receiver].
// ---------------------------------------------------------------------------
__global__ void edge_scatter_kernel(const float* __restrict__ x,
                                    const int*  __restrict__ senders,
                                    const int*  __restrict__ receivers,
                                    float* __restrict__ agg,
                                    float* __restrict__ deg,
                                    int E) {
    long long t = (long long)blockIdx.x * blockDim.x + threadIdx.x;
    long long total = (long long)E * 16;
    if (t >= total) return;
    int e = (int)(t >> 4);
    int q = (int)(t & 15);
    int s = senders[e];
    int r = receivers[e];
    const float4 v = *(const float4*)(x + (size_t)s * 64 + q * 4);
    float* dst = agg + (size_t)r * 64 + q * 4;
    unsafeAtomicAdd(dst + 0, v.x);
    unsafeAtomicAdd(dst + 1, v.y);
    unsafeAtomicAdd(dst + 2, v.z);
    unsafeAtomicAdd(dst + 3, v.w);
    if (q == 0) unsafeAtomicAdd(deg + r, 1.0f);
}

// ---------------------------------------------------------------------------
// Phase 2: per 16-node tile (one wave32 per block):
//   h = (deg*x) @ (Wt-Wb) + agg @ Wb + deg*b_edge     (V_WMMA_F32_16X16X4_F32)
//   h = BN(h);  r = relu(h @ W1 + b1);  out = sigmoid(r @ W2 + b2)
// All WMMA operands staged in fragment-major LDS layout -> pure ds_load_b64.
// ---------------------------------------------------------------------------
__global__ __launch_bounds__(32)
void node_gemm_head_kernel(const float* __restrict__ x,
                           const float* __restrict__ W_edge,  // [128,32]
                           const float* __restrict__ b_edge,  // [32]
                           const float* __restrict__ gamma,
                           const float* __restrict__ beta,
                           const float* __restrict__ mean,
                           const float* __restrict__ var,
                           const float* __restrict__ W1,      // [32,16]
                           const float* __restrict__ b1,      // [16]
                           const float* __restrict__ W2,      // [16]
                           const float* __restrict__ b2,      // [1]
                           const float* __restrict__ agg,     // [N,64]
                           const float* __restrict__ deg,     // [N]
                           float* __restrict__ out, int N) {
    // Fragment-major buffers (dwords): 16 K-steps * 64 for K=64 operands,
    // 8 K-steps * 64 for K=32 operands. Two column tiles for the 32-wide C.
    __shared__ __align__(16) float sWcF[2][16 * 64];  // (Wt - Wb) B-frags, col tiles 0/1
    __shared__ __align__(16) float sWbF[2][16 * 64];  // Wb B-frags
    __shared__ __align__(16) float sXF [16 * 64];     // deg*x A-frags
    __shared__ __align__(16) float sAF [16 * 64];     // agg   A-frags
    __shared__ __align__(16) float sHF [8 * 64];      // BN(h) A-frags (head GEMM)
    __shared__ __align__(16) float sW1F[8 * 64];      // W1 B-frags

    const int lane = threadIdx.x;
    const int n0   = blockIdx.x * 16;
    const int half = lane >> 4;      // 0: K={0,1}, 1: K={2,3} within a K4 step
    const int sub  = lane & 15;      // A row / B,C,D column

    // Stage edge weights into B-fragment layout: Wc = Wt - Wb, Wb.
    for (int i = lane; i < 64 * 32; i += 32) {
        int k = i >> 5, c = i & 31;
        int ct = c >> 4, col = c & 15;
        float wt = W_edge[k * 32 + c];
        float wb = W_edge[(k + 64) * 32 + c];
        int fi = fragIdx(k, col);
        sWcF[ct][fi] = wt - wb;
        sWbF[ct][fi] = wb;
    }
    // Stage W1 (32x16 row-major) into B-fragment layout.
    for (int i = lane; i < 32 * 16; i += 32) {
        int k = i >> 4, col = i & 15;
        sW1F[fragIdx(k, col)] = W1[i];
    }
    // Stage node tiles into A-fragment layout: sXF = deg*x, sAF = agg.
    for (int i = lane; i < 16 * 64; i += 32) {
        int r = i >> 6, k = i & 63;
        int n = n0 + r; if (n >= N) n = N - 1;
        float d = deg[n];
        int fi = fragIdx(k, r);
        sXF[fi] = d * x[(size_t)n * 64 + k];
        sAF[fi] = agg[(size_t)n * 64 + k];
    }
    __syncthreads();

    // EdgeConv GEMM: two 16x64x32 fp32 WMMA GEMMs, C split into two 16-col tiles.
    const int fo = lane * 2;         // per-lane fragment dword offset
    v8f acc0 = {};   // channels 0..15
    v8f acc1 = {};   // channels 16..31
    for (int t = 0; t < 16; ++t) {
        const int o = t * 64 + fo;
        v2f aX  = *(const v2f*)&sXF[o];
        v2f aA  = *(const v2f*)&sAF[o];
        v2f bc0 = *(const v2f*)&sWcF[0][o];
        v2f bc1 = *(const v2f*)&sWcF[1][o];
        v2f bb0 = *(const v2f*)&sWbF[0][o];
        v2f bb1 = *(const v2f*)&sWbF[1][o];
        acc0 = __builtin_amdgcn_wmma_f32_16x16x4_f32(false, aX, false, bc0, (short)0, acc0, false, false);
        acc1 = __builtin_amdgcn_wmma_f32_16x16x4_f32(false, aX, false, bc1, (short)0, acc1, false, false);
        acc0 = __builtin_amdgcn_wmma_f32_16x16x4_f32(false, aA, false, bb0, (short)0, acc0, false, false);
        acc1 = __builtin_amdgcn_wmma_f32_16x16x4_f32(false, aA, false, bb1, (short)0, acc1, false, false);
    }

    // Epilogue 1: + deg*b_edge, BatchNorm (inference); write straight into
    // A-fragment layout for the head GEMM.
    {
        const int c0 = sub, c1 = sub + 16;
        const float sc0 = gamma[c0] * rsqrtf(var[c0] + BN_EPS);
        const float of0 = beta[c0] - sc0 * mean[c0];
        const float be0 = b_edge[c0];
        const float sc1 = gamma[c1] * rsqrtf(var[c1] + BN_EPS);
        const float of1 = beta[c1] - sc1 * mean[c1];
        const float be1 = b_edge[c1];
        for (int r = 0; r < 8; ++r) {
            int m = r + half * 8;
            int n = n0 + m; if (n >= N) n = N - 1;
            float d = deg[n];
            sHF[fragIdx(c0, m)] = sc0 * (acc0[r] + d * be0) + of0;
            sHF[fragIdx(c1, m)] = sc1 * (acc1[r] + d * be1) + of1;
        }
    }
    __syncthreads();

    // Head GEMM: [16 nodes x 32] @ W1[32 x 16] via 8 fp32 WMMA steps.
    v8f acc2 = {};
#pragma unroll
    for (int t = 0; t < 8; ++t) {
        const int o = t * 64 + fo;
        v2f a = *(const v2f*)&sHF[o];
        v2f b = *(const v2f*)&sW1F[o];
        acc2 = __builtin_amdgcn_wmma_f32_16x16x4_f32(false, a, false, b, (short)0, acc2, false, false);
    }

    // Epilogue 2: relu, dot with W2 (16-lane shfl_xor reduction), sigmoid.
    const float b1j = b1[sub];
    const float w2j = W2[sub];
    const float b2v = b2[0];
    for (int r = 0; r < 8; ++r) {
        float v = fmaxf(acc2[r] + b1j, 0.0f) * w2j;
        v += __shfl_xor(v, 8, 32);
        v += __shfl_xor(v, 4, 32);
        v += __shfl_xor(v, 2, 32);
        v += __shfl_xor(v, 1, 32);
        if (sub == 0) {
            int n = n0 + r + half * 8;
            if (n < N) out[n] = 1.0f / (1.0f + expf(-(v + b2v)));
        }
    }
}

// ---------------------------------------------------------------------------
extern "C" void kernel_launch(void* const* d_in, const int* in_sizes, int n_in,
                              void* d_out, int out_size, void* d_ws, size_t ws_size,
                              hipStream_t stream) {
    const float* x         = (const float*)d_in[0];
    const int*   senders   = (const int*)  d_in[1];
    const int*   receivers = (const int*)  d_in[2];
    const float* W_edge    = (const float*)d_in[3];
    const float* b_edge    = (const float*)d_in[4];
    const float* gamma     = (const float*)d_in[5];
    const float* beta      = (const float*)d_in[6];
    const float* mov_mean  = (const float*)d_in[7];
    const float* mov_var   = (const float*)d_in[8];
    const float* W1        = (const float*)d_in[9];
    const float* b1        = (const float*)d_in[10];
    const float* W2        = (const float*)d_in[11];
    const float* b2        = (const float*)d_in[12];
    float* out = (float*)d_out;

    const int N = in_sizes[0] / 64;
    const int E = in_sizes[1];

    float* agg = (float*)d_ws;                    // [N*64]
    float* deg = agg + (size_t)N * 64;            // [N]

    // Phase 0: zero agg+deg (deterministic every call; graph-capture safe).
    {
        long long n = (long long)N * 65;
        int tb = 256;
        int nb = (int)((n + tb - 1) / tb);
        zero_ws_kernel<<<nb, tb, 0, stream>>>((float*)d_ws, n);
    }
    // Phase 1: scatter x[senders] into agg[receivers], count deg.
    {
        long long total = (long long)E * 16;
        int tb = 256;
        int nb = (int)((total + tb - 1) / tb);
        edge_scatter_kernel<<<nb, tb, 0, stream>>>(x, senders, receivers, agg, deg, E);
    }
    // Phase 2: fused WMMA GEMMs + BN + dense head.
    {
        int nblocks = (N + 15) / 16;
        node_gemm_head_kernel<<<nblocks, 32, 0, stream>>>(
            x, W_edge, b_edge, gamma, beta, mov_mean, mov_var,
            W1, b1, W2, b2, agg, deg, out, N);
    }
}